// MatchingMask_75857712382594
// MI455X (gfx1250) — compile-verified
//
#include <hip/hip_runtime.h>

// Batched Hungarian (Jonker-Volgenant) for B x 100 x 100 cost matrices.
// One wave32 per batch element. Each lane owns 4 columns (j = k*32 + lane).
// CDNA5-specific paths: global_load_async_to_lds_b128 staging of the 40KB
// cost tile + s_wait_asynccnt, ds_bpermute-based wave32 shuffles for the
// masked argmin / path bookkeeping.

#define NMAX  100
#define INF_F 1.0e9f

typedef int v4i __attribute__((ext_vector_type(4)));
typedef __attribute__((address_space(1))) v4i as1_v4i;
typedef __attribute__((address_space(3))) v4i as3_v4i;

__device__ __forceinline__ void async_cp16(const void* g, void* l) {
#if __has_builtin(__builtin_amdgcn_global_load_async_to_lds_b128)
  // per-lane 16B async copy from the global address g to the LDS address l; tracked by ASYNCcnt
  __builtin_amdgcn_global_load_async_to_lds_b128(
      (as1_v4i*)(unsigned long long)g,
      (as3_v4i*)(unsigned)(unsigned long long)l,
      /*imm offset*/ 0, /*imm cpol*/ 0);
#else
  *(float4*)l = *(const float4*)g;  // synchronous fallback
#endif
}

__device__ __forceinline__ void wait_async_zero() {
#if __has_builtin(__builtin_amdgcn_s_wait_asynccnt)
  __builtin_amdgcn_s_wait_asynccnt(0);
  asm volatile("" ::: "memory");
#else
  asm volatile("s_wait_asynccnt 0" ::: "memory");
#endif
}

// select one of 4 per-lane registers by a wave-uniform slot index
__device__ __forceinline__ int sel4(const int a[4], int slot) {
  return (slot == 0) ? a[0] : (slot == 1) ? a[1] : (slot == 2) ? a[2] : a[3];
}

__global__ __launch_bounds__(32)
void hungarian_wave32_kernel(const float* __restrict__ cost_array,
                             const int* __restrict__ num_objects,
                             float* __restrict__ mask_out,
                             float* __restrict__ assigned_out) {
  __shared__ float c_lds[NMAX * NMAX];  // 40000 B cost tile
  __shared__ float u_lds[128];          // row potentials (100 used)

  const int b    = blockIdx.x;
  const int lane = threadIdx.x;  // 0..31, wave32
  const float* cost = cost_array + (size_t)b * (NMAX * NMAX);
  const int nobj = num_objects[b];

  // ---- async stage cost tile: 40000 B = 2500 x 16B chunks -----------------
  {
    const char* gs = (const char*)cost;
    char* ls = (char*)c_lds;
    for (int it = 0; it < 78; ++it) {          // 78*32 = 2496 chunks
      int idx = it * 32 + lane;
      async_cp16(gs + (size_t)idx * 16, ls + (size_t)idx * 16);
    }
    if (lane < 4) {                            // tail: chunks 2496..2499
      int idx = 78 * 32 + lane;
      async_cp16(gs + (size_t)idx * 16, ls + (size_t)idx * 16);
    }
  }

  // ---- init state while the async copy is in flight -----------------------
  volatile float* uv = u_lds;
  for (int k = 0; k < 4; ++k) uv[k * 32 + lane] = 0.0f;  // u = 0

  float v_[4];  // column potentials
  int   p_[4];  // p[j]: row matched to column j (-1 = free)
  for (int k = 0; k < 4; ++k) { v_[k] = 0.0f; p_[k] = -1; }

  wait_async_zero();
  __syncthreads();

  // ---- JV: add rows one at a time -----------------------------------------
  for (int r = 0; r < NMAX; ++r) {
    float minv_[4]; int way_[4]; int used_[4];
    for (int k = 0; k < 4; ++k) { minv_[k] = INF_F; way_[k] = -1; used_[k] = 0; }

    int j0 = -1;
    int i0 = r;  // first iteration uses row r
    for (;;) {
      const float u_i0 = uv[i0];
      const int   row_real = (i0 < nobj);
      const float* crow = &c_lds[i0 * NMAX];

      // reduced-cost scan + local argmin candidate (lowest-j tie-break)
      float bv = INF_F;
      int   bj = 0x7fffffff;
#pragma unroll
      for (int k = 0; k < 4; ++k) {
        int j = k * 32 + lane;
        if (j < NMAX) {
          float cij = row_real ? crow[j] : 0.0f;
          float cur = cij - u_i0 - v_[k];
          if (!used_[k] && cur < minv_[k]) { minv_[k] = cur; way_[k] = j0; }
        }
        float cv = (j < NMAX && !used_[k]) ? minv_[k] : INF_F;
        if (cv < bv || (cv == bv && j < bj)) { bv = cv; bj = j; }
      }
      // wave32 lexicographic (value, index) min — matches jnp.argmin ties
#pragma unroll
      for (int off = 16; off >= 1; off >>= 1) {
        float ov = __shfl_xor(bv, off, 32);
        int   oj = __shfl_xor(bj, off, 32);
        if (ov < bv || (ov == bv && oj < bj)) { bv = ov; bj = oj; }
      }
      const float delta = bv;
      const int   j1    = bj;

      // dual updates (all target rows are pairwise distinct -> no atomics)
      if (lane == (r & 31)) uv[r] = uv[r] + delta;
#pragma unroll
      for (int k = 0; k < 4; ++k) {
        int j = k * 32 + lane;
        if (j < NMAX) {
          if (used_[k]) {
            int pi = p_[k];
            if (pi >= 0) uv[pi] = uv[pi] + delta;
            v_[k] -= delta;
          } else {
            minv_[k] -= delta;
          }
        }
      }
      // mark j1 used (owner lane only; slot index is wave-uniform)
      {
        int slot = j1 >> 5;
        if (lane == (j1 & 31)) {
          if (slot == 0) used_[0] = 1; else if (slot == 1) used_[1] = 1;
          else if (slot == 2) used_[2] = 1; else used_[3] = 1;
        }
      }
      j0 = j1;
      // continue while the chosen column is already matched
      int pj0 = __shfl(sel4(p_, j0 >> 5), j0 & 31, 32);
      if (pj0 < 0) break;
      i0 = pj0;
    }

    // augment: flip matched edges back along 'way' to row r
    while (j0 >= 0) {
      int j1 = __shfl(sel4(way_, j0 >> 5), j0 & 31, 32);
      int pi;
      if (j1 < 0) pi = r;
      else        pi = __shfl(sel4(p_, j1 >> 5), j1 & 31, 32);
      int slot = j0 >> 5;
      if (lane == (j0 & 31)) {
        if (slot == 0) p_[0] = pi; else if (slot == 1) p_[1] = pi;
        else if (slot == 2) p_[2] = pi; else p_[3] = pi;
      }
      j0 = j1;
    }
  }

  // ---- outputs -------------------------------------------------------------
  float* mask = mask_out + (size_t)b * (NMAX * NMAX);
  // zero-fill mask tile (16B vector stores), then scatter the ones.
  float4 z4 = make_float4(0.f, 0.f, 0.f, 0.f);
  for (int it = 0; it < 78; ++it) {
    ((float4*)mask)[it * 32 + lane] = z4;
  }
  if (lane < 4) ((float4*)mask)[2496 + lane] = z4;

#pragma unroll
  for (int k = 0; k < 4; ++k) {
    int j = k * 32 + lane;
    if (j < NMAX) {
      int pj = p_[k];
      int hit = (pj >= 0 && pj < nobj);
      assigned_out[(size_t)b * NMAX + j] = hit ? 1.0f : 0.0f;
      if (hit) mask[pj * NMAX + j] = 1.0f;  // same-wave stores stay ordered
    }
  }
}

extern "C" void kernel_launch(void* const* d_in, const int* in_sizes, int n_in,
                              void* d_out, int out_size, void* d_ws, size_t ws_size,
                              hipStream_t stream) {
  (void)n_in; (void)out_size; (void)d_ws; (void)ws_size;
  const float* cost = (const float*)d_in[0];
  const int*   nobj = (const int*)d_in[1];
  const int B = (n_in > 1) ? in_sizes[1] : 256;

  float* mask     = (float*)d_out;                          // [B,100,100]
  float* assigned = mask + (size_t)B * NMAX * NMAX;         // [B,100,1]

  hipLaunchKernelGGL(hungarian_wave32_kernel, dim3(B), dim3(32), 0, stream,
                     cost, nobj, mask, assigned);
}